// TrainGRPOModuleJax_73229192397434
// MI455X (gfx1250) — compile-verified
//
#include <hip/hip_runtime.h>
#include <hip/hip_bf16.h>
#include <stdint.h>

// Problem constants (from reference: B=4, T=257, V=131072)
#define NB        4
#define NT        257
#define NTP       256          // T-1 rows per batch
#define NV        131072
#define WAVES     8
#define PER_WAVE  16384        // NV / WAVES floats per wave
#define SLOT_F    512          // floats per slot per wave (4 x 32-lane b128 issues)
#define NBUF      4            // LDS ring depth per wave
#define NSLOTS    32           // PER_WAVE / SLOT_F

// ---- CDNA5 async global->LDS copy (ASYNCcnt-tracked) ----
__device__ __forceinline__ void async_b128(uint32_t lds_byte, const float* gptr) {
  asm volatile("global_load_async_to_lds_b128 %0, %1, off"
               :: "v"(lds_byte), "v"((uint64_t)(uintptr_t)gptr)
               : "memory");
}
#define S_WAIT_ASYNC(n) asm volatile("s_wait_asynccnt %0" :: "n"(n) : "memory")
#define S_WAIT_DS0()    asm volatile("s_wait_dscnt 0x0" ::: "memory")

// merge two online-softmax partials (m, Z, S) where S = sum x*exp(x-m)
__device__ __forceinline__ void combine(float& m, float& z, float& s,
                                        float m2, float z2, float s2) {
  float M = fmaxf(m, m2);
  float a = __expf(m  - M);
  float b = __expf(m2 - M);
  z = z * a + z2 * b;
  s = s * a + s2 * b;
  m = M;
}

__global__ __launch_bounds__(256)
void grpo_row_kernel(const float* __restrict__ logits,
                     const int*   __restrict__ input_ids,
                     float*       __restrict__ out,
                     float*       __restrict__ ent_ws) {
  __shared__ __align__(16) float buf[WAVES * NBUF * SLOT_F];   // 64 KB ring buffers
  __shared__ float rm[WAVES], rz[WAVES], rs[WAVES];

  const int row  = blockIdx.x;           // 0..1023  (b*256 + t)
  const int b    = row >> 8;
  const int t    = row & 255;
  const size_t row_base = ((size_t)(b * NT + t)) * (size_t)NV;  // skip last timestep
  const int wave = threadIdx.x >> 5;
  const int lane = threadIdx.x & 31;

  const float*   gbase    = logits + row_base + (size_t)wave * PER_WAVE;
  const uint32_t lds_base =
      (uint32_t)(uintptr_t)&buf[wave * NBUF * SLOT_F];   // low 32 bits of flat ptr = LDS offset

  // issue one slot: 4 coalesced b128 async copies (32 lanes x 16B each)
  auto issue = [&](int s) {
    uint32_t l0 = lds_base + (uint32_t)((s & (NBUF - 1)) * SLOT_F * 4) + (uint32_t)(lane * 16);
    const float* g0 = gbase + s * SLOT_F + lane * 4;
#pragma unroll
    for (int j = 0; j < 4; ++j)
      async_b128(l0 + j * 512, g0 + j * 128);
  };

  float m = -INFINITY, z = 0.f, sx = 0.f;

  // Two-phase slot reduction: tree-max, then 16 independent exps (ILP),
  // then ONE running combine per slot (short serial chain).
  auto process_slot = [&](float4 v0, float4 v1, float4 v2, float4 v3) {
    float xs[16] = {v0.x, v0.y, v0.z, v0.w, v1.x, v1.y, v1.z, v1.w,
                    v2.x, v2.y, v2.z, v2.w, v3.x, v3.y, v3.z, v3.w};
    // tree max over 16 values (log2 depth = 4)
    float t8[8];
#pragma unroll
    for (int k = 0; k < 8; ++k) t8[k] = fmaxf(xs[k], xs[k + 8]);
    float t4a = fmaxf(t8[0], t8[4]), t4b = fmaxf(t8[1], t8[5]);
    float t4c = fmaxf(t8[2], t8[6]), t4d = fmaxf(t8[3], t8[7]);
    float sm  = fmaxf(fmaxf(t4a, t4b), fmaxf(t4c, t4d));
    // 16 independent exps, 4-way partial sums for ILP
    float za = 0.f, zb = 0.f, zc = 0.f, zd = 0.f;
    float sa = 0.f, sb = 0.f, sc = 0.f, sd = 0.f;
#pragma unroll
    for (int k = 0; k < 4; ++k) {
      float ea = __expf(xs[4 * k + 0] - sm);
      float eb = __expf(xs[4 * k + 1] - sm);
      float ec = __expf(xs[4 * k + 2] - sm);
      float ed = __expf(xs[4 * k + 3] - sm);
      za += ea; zb += eb; zc += ec; zd += ed;
      sa = fmaf(xs[4 * k + 0], ea, sa);
      sb = fmaf(xs[4 * k + 1], eb, sb);
      sc = fmaf(xs[4 * k + 2], ec, sc);
      sd = fmaf(xs[4 * k + 3], ed, sd);
    }
    combine(m, z, sx, sm, (za + zb) + (zc + zd), (sa + sb) + (sc + sd));
  };

  const float4* lbase =
      (const float4*)&buf[wave * NBUF * SLOT_F];   // float4 view of this wave's ring

  // prologue: fill the ring (16 async ops in flight)
#pragma unroll
  for (int s = 0; s < NBUF; ++s) issue(s);

  // steady state: wait oldest slot, read it, fence LDS reads, re-issue, compute
  for (int s = 0; s < NSLOTS - NBUF; ++s) {
    S_WAIT_ASYNC(12);                               // (4 slots - 1) * 4 ops => slot s landed
    const int fo = (s & (NBUF - 1)) * (SLOT_F / 4) + lane;  // float4 index
    float4 v0 = lbase[fo];
    float4 v1 = lbase[fo + 32];
    float4 v2 = lbase[fo + 64];
    float4 v3 = lbase[fo + 96];
    S_WAIT_DS0();                                   // reads done before slot is overwritten
    issue(s + NBUF);
    process_slot(v0, v1, v2, v3);
  }

  // tail: drain remaining 4 slots
  S_WAIT_ASYNC(0);
  for (int s = NSLOTS - NBUF; s < NSLOTS; ++s) {
    const int fo = (s & (NBUF - 1)) * (SLOT_F / 4) + lane;
    process_slot(lbase[fo], lbase[fo + 32], lbase[fo + 64], lbase[fo + 96]);
  }

  // wave32 butterfly reduction of (m, Z, S)
#pragma unroll
  for (int off = 16; off > 0; off >>= 1) {
    float m2 = __shfl_xor(m,  off, 32);
    float z2 = __shfl_xor(z,  off, 32);
    float s2 = __shfl_xor(sx, off, 32);
    combine(m, z, sx, m2, z2, s2);
  }
  if (lane == 0) { rm[wave] = m; rz[wave] = z; rs[wave] = sx; }
  __syncthreads();

  if (threadIdx.x == 0) {
    float M = rm[0], Z = rz[0], S = rs[0];
#pragma unroll
    for (int w = 1; w < WAVES; ++w) combine(M, Z, S, rm[w], rz[w], rs[w]);
    float logZ = __logf(Z);
    // entropy with first-order eps correction: -sum p*log(p+1e-9) ~= H - V*eps
    float H = M + logZ - S / Z - 1.31072e-4f;
    ent_ws[row] = H;
    int chosen = input_ids[b * NT + t + 1];
    float xc = logits[row_base + (size_t)chosen];
    out[1 + row] = xc - M - logZ;                   // per_token_logps (TEMPERATURE = 1)
  }
}

__global__ __launch_bounds__(256)
void grpo_finalize_kernel(const int*   __restrict__ labels,
                          const float* __restrict__ adv,
                          const float* __restrict__ ent,
                          float*       __restrict__ out) {
  __shared__ int   sv[NTP];
  __shared__ float acc[4];   // mask_sum, H*mask_sum, ecm_sum, H*ecm_sum
  const int t = threadIdx.x;
  float loss_num = 0.f, total_mask = 0.f;   // thread 0 only

  for (int b = 0; b < NB; ++b) {
    int   mask = labels[b * NT + t + 1];
    float H    = ent[b * NTP + t];
    sv[t] = mask;
    if (t < 4) acc[t] = 0.f;
    __syncthreads();

    int cum = 0;
    for (int i = 0; i <= t; ++i) cum += sv[i];      // inclusive cumsum of valid
    int ecm = (mask == 1) && (cum >= 4) && (cum <= 100);

    atomicAdd(&acc[0], (float)mask);                // ds_add_f32
    atomicAdd(&acc[1], H * (float)mask);
    atomicAdd(&acc[2], (float)ecm);
    atomicAdd(&acc[3], H * (float)ecm);
    __syncthreads();

    if (t == 0) {
      float msum = acc[0];
      out[1025 + b] = acc[1] / msum;                // avg_entropy_per_sample
      out[1029 + b] = acc[3] / acc[2];              // avg_entropy_truncated
      // ratio == exp(0) == 1 -> per_token_loss == -adv[b] on valid tokens
      loss_num   += -adv[b] * msum;
      total_mask += msum;
    }
    __syncthreads();
  }
  if (t == 0) out[0] = loss_num / total_mask;
}

extern "C" void kernel_launch(void* const* d_in, const int* in_sizes, int n_in,
                              void* d_out, int out_size, void* d_ws, size_t ws_size,
                              hipStream_t stream) {
  const float* logits    = (const float*)d_in[0];   // (4, 257, 131072) f32
  const float* adv       = (const float*)d_in[1];   // (4,) f32
  const int*   input_ids = (const int*)d_in[2];     // (4, 257) i32
  const int*   labels    = (const int*)d_in[3];     // (4, 257) i32
  float* out    = (float*)d_out;                    // 1 + 1024 + 4 + 4 = 1033 f32
  float* ent_ws = (float*)d_ws;                     // 1024 f32 scratch (token entropies)

  hipLaunchKernelGGL(grpo_row_kernel, dim3(NB * NTP), dim3(256), 0, stream,
                     logits, input_ids, out, ent_ws);
  hipLaunchKernelGGL(grpo_finalize_kernel, dim3(1), dim3(256), 0, stream,
                     labels, adv, ent_ws, out);
}